// KbAttn_79517024518149
// MI455X (gfx1250) — compile-verified
//
#include <hip/hip_runtime.h>

#define HH 128
#define BB 2048
#define LL 431

typedef __attribute__((ext_vector_type(2))) float v2f;
typedef __attribute__((ext_vector_type(8))) float v8f;

// Stage 0: u1[h] = sum_e v[e]*W[e, h]; u2[h] = sum_e v[e]*W[e, 128+h]; c = v.attn_b
// ws layout (floats): [0..127]=u1, [128..255]=u2, [256]=c, [257..2304]=s1
__global__ void KbAttn_stage0(const float* __restrict__ attn_w,
                              const float* __restrict__ attn_b,
                              const float* __restrict__ v,
                              float* __restrict__ ws) {
    int h = threadIdx.x;  // 128 threads
    float a1 = 0.f, a2 = 0.f;
    for (int e = 0; e < HH; ++e) {
        float ve = v[e];
        a1 += ve * attn_w[e * 2 * HH + h];
        a2 += ve * attn_w[e * 2 * HH + HH + h];
    }
    ws[h]      = a1;
    ws[HH + h] = a2;
    if (h == 0) {
        float cc = 0.f;
        for (int e = 0; e < HH; ++e) cc += v[e] * attn_b[e];
        ws[2 * HH] = cc;
    }
}

// Stage 1: s1[b] = dot(hidden[b,:], u1). One wave32 per b, float4 loads + shuffle reduce.
__global__ void KbAttn_stage1(const float* __restrict__ hidden,
                              const float* __restrict__ ws,
                              float* __restrict__ s1) {
    int wave = (blockIdx.x * blockDim.x + threadIdx.x) >> 5;
    int lane = threadIdx.x & 31;
    if (wave >= BB) return;
    const float4* hv = (const float4*)(hidden + (size_t)wave * HH);
    const float4* uv = (const float4*)(ws);  // u1
    float4 a = hv[lane];
    float4 u = uv[lane];
    float d = a.x * u.x + a.y * u.y + a.z * u.z + a.w * u.w;
    for (int off = 16; off > 0; off >>= 1) d += __shfl_down(d, off, 32);
    if (lane == 0) s1[wave] = d;
}

// Stage 2: streaming matvec over k_embedding via V_WMMA_F32_16X16X4_F32.
// Rows flattened r = l*BB + b (natural memory order of k_embedding).
// A (16x4, f32): lane m=lane&15, half=lane>>4 -> A[m, 2*half + {0,1}] = row[kk*4 + 2*half + {0,1}]
// B (4x16, f32): column-replicated u2 chunk  -> B[2*half + {0,1}, n] = u2[kk*4 + 2*half + {0,1}]
// D (16x16): all N columns equal the dot; lane m==0 of each half holds rows half*8 + i in acc[i].
__global__ void KbAttn_stage2(const float* __restrict__ K,
                              const float* __restrict__ ws,
                              const float* __restrict__ s1,
                              float* __restrict__ out) {
    const int lane = threadIdx.x & 31;
    const int waveInBlock = threadIdx.x >> 5;
    const long tile = (long)blockIdx.x * (blockDim.x >> 5) + waveInBlock;
    const long rowBase = tile * 16;
    const int half = lane >> 4;   // 0 or 1
    const int m = lane & 15;

    const float* __restrict__ u2 = ws + HH;
    const float cc = ws[2 * HH];

    const float* __restrict__ arow = K + (rowBase + m) * HH + half * 2;
    const float* __restrict__ brow = u2 + half * 2;

    v8f acc = {0.f, 0.f, 0.f, 0.f, 0.f, 0.f, 0.f, 0.f};
#pragma unroll
    for (int kk = 0; kk < HH / 4; ++kk) {
        v2f a, b;
        a.x = arow[kk * 4 + 0];
        a.y = arow[kk * 4 + 1];
        b.x = brow[kk * 4 + 0];
        b.y = brow[kk * 4 + 1];
        // 8 args: (neg_a, A, neg_b, B, c_mod, C, reuse_a, reuse_b)
        acc = __builtin_amdgcn_wmma_f32_16x16x4_f32(
            false, a, false, b, (short)0, acc, false, false);
    }

    // Lane 0 holds D[0..7,0] in acc[0..7]; lane 16 holds D[8..15,0].
    if (m == 0) {
#pragma unroll
        for (int i = 0; i < 8; ++i) {
            long r = rowBase + half * 8 + i;  // r = l*BB + b
            long l = r >> 11;                 // r / 2048
            long b = r & (BB - 1);            // r % 2048
            out[b * LL + l] = acc[i] + s1[b] + cc;
        }
    }
}

extern "C" void kernel_launch(void* const* d_in, const int* in_sizes, int n_in,
                              void* d_out, int out_size, void* d_ws, size_t ws_size,
                              hipStream_t stream) {
    const float* hidden = (const float*)d_in[0];   // (1, 2048, 128)
    const float* kemb   = (const float*)d_in[1];   // (431, 2048, 128)
    const float* attn_w = (const float*)d_in[2];   // (128, 256)
    const float* attn_b = (const float*)d_in[3];   // (128,)
    const float* v      = (const float*)d_in[4];   // (1, 128)
    float* out = (float*)d_out;                    // (2048, 431)
    float* ws  = (float*)d_ws;
    float* s1  = ws + 2 * HH + 1;

    KbAttn_stage0<<<1, HH, 0, stream>>>(attn_w, attn_b, v, ws);
    KbAttn_stage1<<<(BB * 32) / 256, 256, 0, stream>>>(hidden, ws, s1);

    const long tiles = (long)LL * BB / 16;       // 55168
    const int wavesPerBlock = 8;                 // 256 threads
    KbAttn_stage2<<<(int)(tiles / wavesPerBlock), 256, 0, stream>>>(kemb, ws, s1, out);
}